// VectorQuantizer_90031104459318
// MI455X (gfx1250) — compile-verified
//
#include <hip/hip_runtime.h>
#include <hip/hip_bf16.h>

typedef __attribute__((ext_vector_type(2))) float v2f;
typedef __attribute__((ext_vector_type(8))) float v8f;

#define N_E        1024
#define E_DIM      64
#define N_TOK      262144
#define BETA       0.25f
#define CB_STRIDE  68                      // 64 + 4 pad -> conflict-free ds_load_b64
#define LDS_FLOATS (N_E * CB_STRIDE + N_E) // codebook (padded) + per-row |c|^2

__global__ void vq_zero_ws(float* ws) { ws[0] = 0.0f; }

__global__ __launch_bounds__(256, 1)
void vq_main(const float* __restrict__ x,
             const float* __restrict__ cb,
             float* __restrict__ out_xq,
             float* __restrict__ out_idx,
             float* __restrict__ loss_acc)
{
    extern __shared__ float smem[];
    float* cnorm = smem + N_E * CB_STRIDE;
    const int tid = threadIdx.x;

    // ---- Stage full 256KB codebook into LDS (row stride 68 floats) ----
    for (int it = 0; it < (N_E * E_DIM) / (4 * 256); ++it) {
        int e4     = tid + it * 256;
        int linear = e4 * 4;
        int row    = linear >> 6;
        int col    = linear & 63;
        float4 v = *(const float4*)(cb + linear);
        *(float4*)(smem + row * CB_STRIDE + col) = v;
    }
    __syncthreads();

    // ---- Per-row |c|^2 into LDS ----
    for (int j = 0; j < 4; ++j) {
        int r = tid * 4 + j;
        const float* rp = smem + r * CB_STRIDE;
        float s = 0.0f;
#pragma unroll
        for (int c = 0; c < E_DIM; ++c) { float v = rp[c]; s += v * v; }
        cnorm[r] = s;
    }
    __syncthreads();

    const int lane = tid & 31;
    const int wave = tid >> 5;
    const int n    = lane & 15;   // N (code) / M (token) position within half
    const int h    = lane >> 4;   // K-pair select per WMMA f32 layout

    for (int i = 0; i < 8; ++i) {
        const int tile = (blockIdx.x * 8 + wave) * 8 + i;   // 16-token tile
        const int row0 = tile * 16;

        // ---- Load A = -2*x : 16 K-slices of 16x4 f32 ----
        v2f a[16];
        const float* xr = x + (size_t)(row0 + n) * E_DIM + 2 * h;
#pragma unroll
        for (int s = 0; s < 16; ++s) {
            float2 t2 = *(const float2*)(xr + 4 * s);
            a[s].x = -2.0f * t2.x;
            a[s].y = -2.0f * t2.y;
        }

        float bestv[8];
        int   besti[8];
#pragma unroll
        for (int r = 0; r < 8; ++r) { bestv[r] = 3.4e38f; besti[r] = 0; }

        // ---- Sweep 64 code tiles: 16 chained f32 WMMAs each ----
        for (int t = 0; t < 64; ++t) {
            v8f d = {0.f, 0.f, 0.f, 0.f, 0.f, 0.f, 0.f, 0.f};
            const float* bp = smem + (t * 16 + n) * CB_STRIDE + 2 * h;
#pragma unroll
            for (int s = 0; s < 16; ++s) {
                float2 b2 = *(const float2*)(bp + 4 * s);
                v2f b; b.x = b2.x; b.y = b2.y;
                d = __builtin_amdgcn_wmma_f32_16x16x4_f32(
                        false, a[s], false, b, (short)0, d, false, false);
            }
            const float cn = cnorm[t * 16 + n];
            const int   ci = t * 16 + n;
#pragma unroll
            for (int r = 0; r < 8; ++r) {
                float dist   = d[r] + cn;        // |c|^2 - 2 x.c
                bool  better = dist < bestv[r];
                bestv[r] = better ? dist : bestv[r];
                besti[r] = better ? ci   : besti[r];
            }
        }

        // ---- Argmin reduction across the 16 lanes of each half ----
#pragma unroll
        for (int m = 1; m < 16; m <<= 1) {
#pragma unroll
            for (int r = 0; r < 8; ++r) {
                float ov = __shfl_xor(bestv[r], m, 32);
                int   oi = __shfl_xor(besti[r], m, 32);
                if (ov < bestv[r] || (ov == bestv[r] && oi < besti[r])) {
                    bestv[r] = ov; besti[r] = oi;
                }
            }
        }

        // lanes 0..15 hold rows 0..7 (per VGPR), lanes 16..31 rows 8..15
        if (lane == 0) {
            for (int r = 0; r < 8; ++r) out_idx[row0 + r] = (float)besti[r];
        } else if (lane == 16) {
            for (int r = 0; r < 8; ++r) out_idx[row0 + 8 + r] = (float)besti[r];
        }

        // ---- Gather x_q from LDS codebook, write out, accumulate loss ----
        float lsum = 0.0f;
#pragma unroll
        for (int m = 0; m < 16; ++m) {
            int idx = (m < 8) ? __shfl(besti[m], 0, 32)
                              : __shfl(besti[m - 8], 16, 32);
            int row = row0 + m;
            float2 xv = *(const float2*)(x    + (size_t)row * E_DIM + lane * 2);
            float2 cv = *(const float2*)(smem + idx * CB_STRIDE     + lane * 2);
            float dx = cv.x - xv.x, dy = cv.y - xv.y;
            lsum += dx * dx + dy * dy;
            *(float2*)(out_xq + (size_t)row * E_DIM + lane * 2) = cv;
        }
#pragma unroll
        for (int m = 16; m >= 1; m >>= 1) lsum += __shfl_xor(lsum, m, 32);
        if (lane == 0) atomicAdd(loss_acc, lsum);
    }
}

__global__ void vq_finish(const float* ws, float* out_loss) {
    out_loss[0] = ws[0] * ((1.0f + BETA) / (float)((size_t)N_TOK * E_DIM));
}

extern "C" void kernel_launch(void* const* d_in, const int* in_sizes, int n_in,
                              void* d_out, int out_size, void* d_ws, size_t ws_size,
                              hipStream_t stream) {
    const float* x  = (const float*)d_in[0];
    const float* cb = (const float*)d_in[1];

    float* out      = (float*)d_out;
    float* out_xq   = out;                            // [N_TOK*E_DIM]
    float* out_loss = out + (size_t)N_TOK * E_DIM;    // [1]
    float* out_idx  = out_loss + 1;                   // [N_TOK] (as float)
    float* ws       = (float*)d_ws;

    vq_zero_ws<<<1, 1, 0, stream>>>(ws);

    const size_t lds_bytes = (size_t)LDS_FLOATS * sizeof(float); // ~276 KB < 320 KB
    vq_main<<<dim3(256), dim3(256), lds_bytes, stream>>>(x, cb, out_xq, out_idx, ws);

    vq_finish<<<1, 1, 0, stream>>>(ws, out_loss);
}